// GraphSAGE_43078521979013
// MI455X (gfx1250) — compile-verified
//
#include <hip/hip_runtime.h>

// GraphSAGE 3-layer forward for MI455X (gfx1250, wave32).
// Aggregation = coalesced f32 L2 atomics; GEMMs = V_WMMA_F32_16X16X4_F32 (exact f32).

#define NNODES 50000
#define NEDGES 800000
#define KDIM   128   // feature dim of every layer input

typedef __attribute__((ext_vector_type(2))) float v2f;
typedef __attribute__((ext_vector_type(8))) float v8f;

// ---------------- degree: deg[dst[e]] += 1 ----------------
__global__ void sage_degree_kernel(const int* __restrict__ dst,
                                   float* __restrict__ deg, int E) {
    int e = blockIdx.x * blockDim.x + threadIdx.x;
    if (e < E) atomicAdd(&deg[dst[e]], 1.0f);
}

// ---------------- scatter: agg[dst[e],:] += h[src[e],:]  (dim=128) ----------
// One edge per wave32. Lane L handles dims {L, L+32, L+64, L+96}:
// each of the 4 load / atomic-add instructions is a contiguous 128B per wave.
__global__ void sage_scatter_kernel(const float* __restrict__ h,
                                    const int* __restrict__ src,
                                    const int* __restrict__ dst,
                                    float* __restrict__ agg, int E) {
    int wid  = threadIdx.x >> 5;
    int lane = threadIdx.x & 31;
    int e = blockIdx.x * (blockDim.x >> 5) + wid;
    if (e >= E) return;
    int s = src[e];
    int d = dst[e];
    const float* hp = h   + (long)s * KDIM + lane;
    float*       ap = agg + (long)d * KDIM + lane;
#pragma unroll
    for (int k = 0; k < 4; ++k)
        atomicAdd(ap + k * 32, hp[k * 32]);
}

// ---------------- fused SAGE layer GEMM -------------------------------------
// out[r, c] = act( h[r,:]·Ws[:,c] + (agg[r,:]/max(deg[r],1))·Wn[:,c] + b[c] )
// One 16x16 output tile per wave; K walked in steps of 4 via f32 WMMA.
//
// A fragment (16x4, f32, ISA 7.12.2): lane L<16 -> row M=L, K={k,k+1};
//                                     lane L>=16 -> row M=L-16, K={k+2,k+3}.
// B fragment (4x16): lane L<16 -> K=k+{0,1}, N=L; lane L>=16 -> K=k+{2,3}, N=L-16.
// C/D (16x16): VGPR v, lanes 0-15 -> M=v, lanes 16-31 -> M=v+8; N=lane&15.
template <int NC, bool RELU>
__global__ void sage_gemm_kernel(const float* __restrict__ h,
                                 const float* __restrict__ agg,
                                 const float* __restrict__ deg,
                                 const float* __restrict__ Ws,   // [KDIM][NC] row-major
                                 const float* __restrict__ Wn,   // [KDIM][NC] row-major
                                 const float* __restrict__ bias, // [NC]
                                 float* __restrict__ out) {      // [NNODES][NC]
    constexpr int colTiles = NC >> 4;
    constexpr int nTiles   = (NNODES / 16) * colTiles;

    int tile = blockIdx.x * (blockDim.x >> 5) + (threadIdx.x >> 5);
    if (tile >= nTiles) return;              // wave-uniform -> EXEC stays all-ones

    int lane = threadIdx.x & 31;
    int r0 = (tile / colTiles) << 4;
    int c0 = (tile % colTiles) << 4;

    int m    = lane & 15;
    int koff = (lane >> 4) << 1;             // 0 for lanes 0-15, 2 for lanes 16-31
    int row  = r0 + m;
    int n    = c0 + m;

    float idg = 1.0f / fmaxf(deg[row], 1.0f);

    const float* hrow = h   + (long)row * KDIM + koff;
    const float* grow = agg + (long)row * KDIM + koff;
    const float* wsp  = Ws + koff * NC + n;
    const float* wnp  = Wn + koff * NC + n;

    v8f acc = {0.f, 0.f, 0.f, 0.f, 0.f, 0.f, 0.f, 0.f};

#pragma unroll 8
    for (int k0 = 0; k0 < KDIM; k0 += 4) {
        v2f a1, a2, b1, b2;
        a1.x = hrow[k0];
        a1.y = hrow[k0 + 1];
        a2.x = grow[k0] * idg;
        a2.y = grow[k0 + 1] * idg;
        b1.x = wsp[k0 * NC];
        b1.y = wsp[(k0 + 1) * NC];
        b2.x = wnp[k0 * NC];
        b2.y = wnp[(k0 + 1) * NC];
        acc = __builtin_amdgcn_wmma_f32_16x16x4_f32(false, a1, false, b1,
                                                    (short)0, acc, false, false);
        acc = __builtin_amdgcn_wmma_f32_16x16x4_f32(false, a2, false, b2,
                                                    (short)0, acc, false, false);
    }

    float bv    = bias[n];
    int   rbase = r0 + ((lane >> 4) << 3);   // +8 rows for upper half-wave
#pragma unroll
    for (int v = 0; v < 8; ++v) {
        float x = acc[v] + bv;
        if (RELU) x = fmaxf(x, 0.0f);
        out[(long)(rbase + v) * NC + n] = x;
    }
}

// ---------------------------------------------------------------------------
extern "C" void kernel_launch(void* const* d_in, const int* in_sizes, int n_in,
                              void* d_out, int out_size, void* d_ws, size_t ws_size,
                              hipStream_t stream) {
    (void)in_sizes; (void)n_in; (void)out_size; (void)ws_size;

    const float* x   = (const float*)d_in[0];
    const int*   src = (const int*)d_in[1];
    const int*   dst = (const int*)d_in[2];
    const float* Ws0 = (const float*)d_in[3];
    const float* Wn0 = (const float*)d_in[4];
    const float* b0  = (const float*)d_in[5];
    const float* Ws1 = (const float*)d_in[6];
    const float* Wn1 = (const float*)d_in[7];
    const float* b1  = (const float*)d_in[8];
    const float* Ws2 = (const float*)d_in[9];
    const float* Wn2 = (const float*)d_in[10];
    const float* b2  = (const float*)d_in[11];
    float* out = (float*)d_out;

    const size_t feat = (size_t)NNODES * KDIM;
    float* agg = (float*)d_ws;      // [NNODES][128]
    float* hA  = agg + feat;        // [NNODES][128]  layer-0 output
    float* hB  = hA + feat;         // [NNODES][128]  layer-1 output
    float* deg = hB + feat;         // [NNODES]

    const int scatterBlocks = (NEDGES + 7) / 8;          // 8 edges / 256-thr block
    const int gemmBlocks128 = ((NNODES / 16) * 8 + 7) / 8;   // 25000 tiles -> 3125
    const int gemmBlocks64  = ((NNODES / 16) * 4 + 7) / 8;   // 12500 tiles -> 1563

    // degree (shared by all layers)
    hipMemsetAsync(deg, 0, NNODES * sizeof(float), stream);
    sage_degree_kernel<<<(NEDGES + 255) / 256, 256, 0, stream>>>(dst, deg, NEDGES);

    // ---- layer 0: x -> hA (relu) ----
    hipMemsetAsync(agg, 0, feat * sizeof(float), stream);
    sage_scatter_kernel<<<scatterBlocks, 256, 0, stream>>>(x, src, dst, agg, NEDGES);
    sage_gemm_kernel<128, true><<<gemmBlocks128, 256, 0, stream>>>(
        x, agg, deg, Ws0, Wn0, b0, hA);

    // ---- layer 1: hA -> hB (relu) ----
    hipMemsetAsync(agg, 0, feat * sizeof(float), stream);
    sage_scatter_kernel<<<scatterBlocks, 256, 0, stream>>>(hA, src, dst, agg, NEDGES);
    sage_gemm_kernel<128, true><<<gemmBlocks128, 256, 0, stream>>>(
        hA, agg, deg, Ws1, Wn1, b1, hB);

    // ---- layer 2: hB -> out (64 classes, no relu) ----
    hipMemsetAsync(agg, 0, feat * sizeof(float), stream);
    sage_scatter_kernel<<<scatterBlocks, 256, 0, stream>>>(hB, src, dst, agg, NEDGES);
    sage_gemm_kernel<64, false><<<gemmBlocks64, 256, 0, stream>>>(
        hB, agg, deg, Ws2, Wn2, b2, out);
}